// BidirectionalBoxPool_64879775973581
// MI455X (gfx1250) — compile-verified
//
#include <hip/hip_runtime.h>
#include <stdint.h>

#define PH 8

namespace {
constexpr int kN = 8;
constexpr int kK = 64;
constexpr int kC = 128;
constexpr int kH = 128;
constexpr int kW = 128;
constexpr int kCB = 8;          // channels staged/processed per block
constexpr int kWinWords = 1536; // per-channel LDS window capacity (floats)
constexpr int kBlock = 256;
}

typedef __attribute__((ext_vector_type(4))) unsigned int u32x4;
typedef __attribute__((ext_vector_type(8))) unsigned int u32x8;

struct BoxMeta {
  float xmin, ymin, bw, bh;
  int width;
  int wide;
};

// ---------------------------------------------------------------------------
// Kernel 1: per-box metadata + widths output (N,K,2)
// ---------------------------------------------------------------------------
__global__ void bbp_meta_kernel(const float* __restrict__ boxes,
                                float* __restrict__ widths_out,
                                BoxMeta* __restrict__ meta) {
  int b = blockIdx.x * blockDim.x + threadIdx.x;
  if (b >= kN * kK) return;
  float x0 = boxes[b * 4 + 0];
  float y0 = boxes[b * 4 + 1];
  float x1 = boxes[b * 4 + 2];
  float y1 = boxes[b * 4 + 3];
  bool valid = !((x0 == 0.0f) && (y0 == 0.0f) && (x1 == 0.0f) && (y1 == 0.0f));
  float bw = valid ? (x1 - x0) : 1.0f;
  float bh = valid ? (y1 - y0) : 1.0f;
  bool wide = bw > bh;
  float ratio = wide ? (bw / bh) : (bh / bw);
  int width = valid ? (int)ceilf(ratio * (float)PH) : 0;

  BoxMeta m;
  m.xmin = x0; m.ymin = y0; m.bw = bw; m.bh = bh;
  m.width = width; m.wide = wide ? 1 : 0;
  meta[b] = m;

  float wfl = (float)width;
  widths_out[b * 2 + 0] = wfl;
  widths_out[b * 2 + 1] = wfl;
}

// ---------------------------------------------------------------------------
// Kernel 2: bilinear pooling. Block = (box, channel-chunk).
//   - TDM (tensor_load_to_lds) stages the clamped 3D window
//     (Wd x Hd x kCB channels) of x into LDS as one descriptor-driven DMA
//   - compute d=0 samples from LDS, store to d=0 and mirrored d=1 (NT stores)
// ---------------------------------------------------------------------------
__global__ __launch_bounds__(kBlock)
void bbp_pool_kernel(const float* __restrict__ x,
                     const BoxMeta* __restrict__ meta,
                     float* __restrict__ feats,
                     int MW) {
  __shared__ float smem[kCB * kWinWords];

  const int b = blockIdx.x;            // n*K + k
  const int cbase = blockIdx.y * kCB;  // first channel of this block
  const int n = b / kK;
  const int tid = threadIdx.x;

  const BoxMeta m = meta[b];
  const int width = m.width;
  const bool wide = (m.wide != 0);
  const float xmin = m.xmin, ymin = m.ymin, bw = m.bw, bh = m.bh;
  const float wf = fmaxf((float)width, 2.0f);
  const float invw = (width > 0) ? (1.0f / (wf - 1.0f)) : 0.0f;
  const float inv7 = 1.0f / (float)(PH - 1);

  // ---- sampling window (uniform per block), clamped to the image ----
  int xlo = 0, ylo = 0, Wd = 1, Hd = 1;
  bool useLds = false;
  if (width > 0) {
    const float xmax = xmin + bw;
    xlo = max(0, (int)floorf(xmin - 0.5f));
    int xhi = min(kW - 1, (int)floorf(xmax - 0.5f) + 1);
    // tall grids overshoot: y goes up to ymin + wf*bh/(wf-1)
    float ytop = wide ? (ymin + bh - 0.5f) : (ymin + wf * bh * invw - 0.5f);
    ylo = max(0, (int)floorf(ymin - 0.5f));
    int yhi = min(kH - 1, (int)floorf(ytop) + 1);
    Wd = max(1, xhi - xlo + 1);
    Hd = max(1, yhi - ylo + 1);
    useLds = (Hd * Wd) <= kWinWords;

    if (useLds && (tid < 32)) {  // one wave issues the TDM (EXEC-agnostic DMA)
      // ---- build Tensor DMA Descriptor (D#), all block-uniform ----
      unsigned long long gaddr = (unsigned long long)(const void*)(
          x + ((((size_t)n * kC + (size_t)cbase) * kH + (size_t)ylo) * kW +
               (size_t)xlo));
      uint32_t lds = (uint32_t)(uintptr_t)(&smem[0]);
      uint32_t wd = (uint32_t)Wd, hd = (uint32_t)Hd, cb = (uint32_t)kCB;
      uint32_t s0 = (uint32_t)kW;            // dim0 stride (elements)
      uint32_t s1 = (uint32_t)(kH * kW);     // dim1 stride (elements)

      u32x4 g0;
      g0.x = 1u;                                   // count=1 valid descriptor
      g0.y = lds;                                  // lds_addr (bytes)
      g0.z = (uint32_t)(gaddr & 0xffffffffull);    // global_addr[31:0]
      g0.w = (uint32_t)((gaddr >> 32) & 0x01ffffffull) | 0x80000000u; // [56:32] | type=2

      u32x8 g1;
      g1.s0 = (2u << 16);                          // data_size = 4 bytes
      g1.s1 = (wd & 0xffffu) << 16;                // tensor_dim0[15:0]
      g1.s2 = (wd >> 16) | ((hd & 0xffffu) << 16); // tensor_dim0[31:16] | tensor_dim1[15:0]
      g1.s3 = (hd >> 16) | (wd << 16);             // tensor_dim1[31:16] | tile_dim0
      g1.s4 = hd | (cb << 16);                     // tile_dim1 | tile_dim2
      g1.s5 = s0;                                  // tensor_dim0_stride[31:0]
      g1.s6 = (s0 >> 32 & 0u) | ((s1 & 0xffffu) << 16); // stride0[47:32]=0 | stride1[15:0]
      g1.s7 = (s1 >> 16);                          // tensor_dim1_stride[47:16]

      u32x4 g2;
      g2.x = cb;   // tensor_dim2
      g2.y = 1u;   // tensor_dim3 (unused, tile_dim3=0)
      g2.z = 0u;   // tensor_dim2_stride lo (unused)
      g2.w = 0u;   // tensor_dim2_stride hi | tile_dim3=0

      u32x4 g3;
      g3.x = 0u; g3.y = 0u; g3.z = 0u; g3.w = 0u;  // dim3 stride / dim4 / tile_dim4

      asm volatile("tensor_load_to_lds %0, %1, %2, %3"
                   :: "s"(g0), "s"(g1), "s"(g2), "s"(g3)
                   : "memory");
      __builtin_amdgcn_s_wait_tensorcnt(0);
    }
  }
  __syncthreads();

  // ---- compute & store ----
  const size_t dStride = (size_t)kC * PH * (size_t)MW; // one d-slice
  float* out0 = feats + (size_t)b * 2 * dStride;       // d = 0
  float* out1 = out0 + dStride;                        // d = 1

  const int win = Hd * Wd;
  const int elems = kCB * PH * MW;

  for (int e = tid; e < elems; e += kBlock) {
    int j = e % MW;
    int t = e / MW;
    int i = t % PH;
    int cl = t / PH;
    int c = cbase + cl;
    size_t o0 = ((size_t)c * PH + (size_t)i) * (size_t)MW + (size_t)j;

    if (j < width) {
      float ix, iy;
      if (wide) {
        ix = xmin + (float)j * bw * invw - 0.5f;
        iy = ymin + (float)i * bh * inv7 - 0.5f;
      } else {
        ix = xmin + (float)i * bw * inv7 - 0.5f;
        iy = ymin + (wf - (float)j) * bh * invw - 0.5f;
      }
      float xf = floorf(ix), yf = floorf(iy);
      float fx = ix - xf, fy = iy - yf;
      int x0i = (int)xf, y0i = (int)yf;
      float w00 = (1.0f - fx) * (1.0f - fy);
      float w10 = fx * (1.0f - fy);
      float w01 = (1.0f - fx) * fy;
      float w11 = fx * fy;

      auto tap = [&](int xc, int yc) -> float {
        if (xc < 0 || xc > kW - 1 || yc < 0 || yc > kH - 1) return 0.0f;
        if (useLds) {
          return smem[cl * win + (yc - ylo) * Wd + (xc - xlo)];
        }
        return x[(((size_t)n * kC + (size_t)c) * kH + (size_t)yc) * kW + (size_t)xc];
      };

      float s = tap(x0i, y0i) * w00 + tap(x0i + 1, y0i) * w10 +
                tap(x0i, y0i + 1) * w01 + tap(x0i + 1, y0i + 1) * w11;

      __builtin_nontemporal_store(s, &out0[o0]);
      // mirror identity: d1[i,j] = d0[PH-1-i, width-1-j]
      size_t o1 = ((size_t)c * PH + (size_t)(PH - 1 - i)) * (size_t)MW +
                  (size_t)(width - 1 - j);
      __builtin_nontemporal_store(s, &out1[o1]);
    } else {
      // masked columns (and fully-invalid boxes) are zero in both grids
      __builtin_nontemporal_store(0.0f, &out0[o0]);
      __builtin_nontemporal_store(0.0f, &out1[o0]);
    }
  }
}

// ---------------------------------------------------------------------------
// launch
// ---------------------------------------------------------------------------
extern "C" void kernel_launch(void* const* d_in, const int* in_sizes, int n_in,
                              void* d_out, int out_size, void* d_ws, size_t ws_size,
                              hipStream_t stream) {
  (void)in_sizes; (void)n_in; (void)ws_size;
  const float* x = (const float*)d_in[0];
  const float* boxes = (const float*)d_in[1];
  float* out = (float*)d_out;

  // out_size = N*K*2*C*PH*MW + N*K*2  ->  recover static, data-dependent MW
  const size_t perMW = (size_t)kN * kK * 2 * kC * PH; // 1,048,576
  const size_t wcount = (size_t)kN * kK * 2;          // 1,024
  int MW = (int)(((size_t)out_size - wcount) / perMW);
  if (MW < 1) MW = 1;

  float* widths = out + perMW * (size_t)MW;
  BoxMeta* meta = (BoxMeta*)d_ws;

  bbp_meta_kernel<<<dim3((kN * kK + kBlock - 1) / kBlock), dim3(kBlock), 0, stream>>>(
      boxes, widths, meta);

  bbp_pool_kernel<<<dim3(kN * kK, kC / kCB), dim3(kBlock), 0, stream>>>(
      x, meta, out, MW);
}